// GraphConv_m_2018634629392
// MI455X (gfx1250) — compile-verified
//
#include <hip/hip_runtime.h>
#include <hip/hip_bf16.h>
#include <math.h>

typedef __attribute__((ext_vector_type(2))) float v2f;
typedef __attribute__((ext_vector_type(8))) float v8f;

#define BB 64
#define NN 1024
#define FF 64
#define DEG 8
#define LRELU_ALPHA 0.2f
#define MASK_VAL -9.0e15f
#define WT_STRIDE 66   // padded stride for transposed W in LDS (keeps 8B align, avoids bank conflicts)

// -------------------------------------------------------------------------
// K1: h = atoms @ W  via V_WMMA_F32_16X16X4_F32 (full fp32 precision).
// Block = 256 threads = 8 waves; each wave computes a 16x64 tile of h.
// W is staged TRANSPOSED in LDS so each lane's B-fragment (k, k+1 for a
// fixed output column) is contiguous -> single ds_load_b64 into an aligned
// VGPR pair, no register-shuffle movs before the WMMA.
// -------------------------------------------------------------------------
__global__ __launch_bounds__(256) void gat_gemm_h(const float* __restrict__ atoms,
                                                  const float* __restrict__ W,
                                                  float* __restrict__ h) {
    __shared__ float ldsWt[FF * WT_STRIDE];     // Wt[n][k] = W[k][n]
    for (int i = threadIdx.x; i < FF * FF; i += 256) {
        const int kk = i >> 6;                  // row of W (K)
        const int nn = i & 63;                  // col of W (N)
        ldsWt[nn * WT_STRIDE + kk] = W[i];
    }
    __syncthreads();

    const int lane = threadIdx.x & 31;
    const int wave = threadIdx.x >> 5;
    const int l16  = lane & 15;
    const int half = lane >> 4;                 // 0: lanes 0-15, 1: lanes 16-31
    const int rowBase = blockIdx.x * 128 + wave * 16;

    v8f acc0 = {}, acc1 = {}, acc2 = {}, acc3 = {};

    const float* aRow = atoms + (size_t)(rowBase + l16) * FF;
    const float* bCol0 = ldsWt + (l16 +  0) * WT_STRIDE;
    const float* bCol1 = ldsWt + (l16 + 16) * WT_STRIDE;
    const float* bCol2 = ldsWt + (l16 + 32) * WT_STRIDE;
    const float* bCol3 = ldsWt + (l16 + 48) * WT_STRIDE;

    for (int k = 0; k < FF; k += 4) {
        const int kb = k + half * 2;            // per-lane K pair (ISA 16x4 A/B layout)
        // A fragment: A[M = rowBase+l16][kb, kb+1]  (8B-aligned float2 load)
        v2f afrag = *(const v2f*)(aRow + kb);
        // B fragments: Wt[col][kb, kb+1] -> contiguous ds_load_b64
        v2f b0 = *(const v2f*)(bCol0 + kb);
        v2f b1 = *(const v2f*)(bCol1 + kb);
        v2f b2 = *(const v2f*)(bCol2 + kb);
        v2f b3 = *(const v2f*)(bCol3 + kb);

        acc0 = __builtin_amdgcn_wmma_f32_16x16x4_f32(false, afrag, false, b0,
                                                     (short)0, acc0, false, false);
        acc1 = __builtin_amdgcn_wmma_f32_16x16x4_f32(false, afrag, false, b1,
                                                     (short)0, acc1, false, false);
        acc2 = __builtin_amdgcn_wmma_f32_16x16x4_f32(false, afrag, false, b2,
                                                     (short)0, acc2, false, false);
        acc3 = __builtin_amdgcn_wmma_f32_16x16x4_f32(false, afrag, false, b3,
                                                     (short)0, acc3, false, false);
    }

    // C/D layout: VGPR r holds D[M=r][N=lane] (lanes 0-15) / D[M=r+8][N=lane-16]
    #pragma unroll
    for (int r = 0; r < 8; ++r) {
        const int row = rowBase + half * 8 + r;
        float* dst = h + (size_t)row * FF + l16;
        dst[0]  = acc0[r];
        dst[16] = acc1[r];
        dst[32] = acc2[r];
        dst[48] = acc3[r];
    }
}

// -------------------------------------------------------------------------
// K2: s1[row] = h[row,:] . a[:64],  s2[row] = h[row,:] . a[64:128]
// One wave32 per row, shfl_xor tree reduction.
// -------------------------------------------------------------------------
__global__ __launch_bounds__(256) void gat_scores(const float* __restrict__ h,
                                                  const float* __restrict__ a,
                                                  float* __restrict__ s1,
                                                  float* __restrict__ s2) {
    const int lane = threadIdx.x & 31;
    const int wave = threadIdx.x >> 5;
    const int row  = blockIdx.x * 8 + wave;

    const float* hr = h + (size_t)row * FF;
    const float x0 = hr[lane];
    const float x1 = hr[lane + 32];

    float p1 = x0 * a[lane]      + x1 * a[lane + 32];
    float p2 = x0 * a[FF + lane] + x1 * a[FF + lane + 32];

    #pragma unroll
    for (int m = 16; m >= 1; m >>= 1) {
        p1 += __shfl_xor(p1, m, 32);
        p2 += __shfl_xor(p2, m, 32);
    }
    if (lane == 0) {
        s1[row] = p1;
        s2[row] = p2;
    }
}

// -------------------------------------------------------------------------
// K3: per-row sparse softmax over unique neighbors + weighted gather + ELU.
// 64 threads per row (4 rows per 256-thread block). Duplicate edge indices
// are suppressed (boolean adjacency in the reference dedups them).
// -------------------------------------------------------------------------
__global__ __launch_bounds__(256) void gat_aggregate(const float* __restrict__ h,
                                                     const int* __restrict__ edges,
                                                     const float* __restrict__ s1,
                                                     const float* __restrict__ s2,
                                                     float* __restrict__ out) {
    const int grp = threadIdx.x >> 6;        // row group within block: 0..3
    const int t   = threadIdx.x & 63;        // feature index
    const int row = blockIdx.x * 4 + grp;    // flat row in [0, B*N)
    const int batchRow0 = row & ~(NN - 1);   // first row of this batch

    __shared__ float sh_e[4][DEG];
    __shared__ int   sh_j[4][DEG];

    if (t < DEG) {
        const int* er = edges + (size_t)row * DEG;
        const int j = er[t];
        bool dup = false;
        for (int dd = 0; dd < t; ++dd) dup = dup || (er[dd] == j);
        float e;
        if (dup) {
            e = MASK_VAL;                               // exp underflows to 0
        } else {
            const float v = s1[row] + s2[batchRow0 + j];
            e = v > 0.0f ? v : LRELU_ALPHA * v;          // LeakyReLU(0.2)
        }
        sh_e[grp][t] = e;
        sh_j[grp][t] = j;
    }
    __syncthreads();

    // softmax over the (<=8) unmasked entries; MASK_VAL entries -> weight 0
    float m = sh_e[grp][0];
    #pragma unroll
    for (int d = 1; d < DEG; ++d) m = fmaxf(m, sh_e[grp][d]);

    float w[DEG];
    float sum = 0.0f;
    #pragma unroll
    for (int d = 0; d < DEG; ++d) {
        w[d] = expf(sh_e[grp][d] - m);
        sum += w[d];
    }
    const float inv = 1.0f / sum;

    const size_t batchBase = (size_t)batchRow0 * FF;
    float acc = 0.0f;
    #pragma unroll
    for (int d = 0; d < DEG; ++d) {
        acc += (w[d] * inv) * h[batchBase + (size_t)sh_j[grp][d] * FF + t];
    }

    // ELU (alpha = 1)
    const float y = acc > 0.0f ? acc : (expf(acc) - 1.0f);
    out[(size_t)row * FF + t] = y;
}

// -------------------------------------------------------------------------
// Launch
// Inputs (setup_inputs order): atoms f32 (B,N,64), edges int (B,N,8),
//                              W f32 (64,64), a f32 (128,1)
// Workspace layout: h [65536*64 f32] | s1 [65536 f32] | s2 [65536 f32]
// -------------------------------------------------------------------------
extern "C" void kernel_launch(void* const* d_in, const int* in_sizes, int n_in,
                              void* d_out, int out_size, void* d_ws, size_t ws_size,
                              hipStream_t stream) {
    (void)in_sizes; (void)n_in; (void)out_size; (void)ws_size;

    const float* atoms = (const float*)d_in[0];
    const int*   edges = (const int*)d_in[1];
    const float* W     = (const float*)d_in[2];
    const float* a     = (const float*)d_in[3];
    float* out = (float*)d_out;

    const int ROWS = BB * NN;                 // 65536
    float* h  = (float*)d_ws;                 // ROWS * FF
    float* s1 = h + (size_t)ROWS * FF;        // ROWS
    float* s2 = s1 + ROWS;                    // ROWS

    gat_gemm_h   <<<ROWS / 128, 256, 0, stream>>>(atoms, W, h);
    gat_scores   <<<ROWS / 8,   256, 0, stream>>>(h, a, s1, s2);
    gat_aggregate<<<ROWS / 4,   256, 0, stream>>>(h, edges, s1, s2, out);
}